// ImplicitComplexModalFilter_481036337252
// MI455X (gfx1250) — compile-verified
//
#include <hip/hip_runtime.h>
#include <hip/hip_bf16.h>

// Problem constants (match reference)
#define H_     1024
#define N_     64
#define NSSM_  1024
#define L_     2048
#define LHI_   (L_ / 16)      // 128
#define PROWS  64             // lhi rows handled per block (2 blocks per h)
#define PSTRIDE 68            // LDS row stride in floats (bank-conflict free)

typedef float v2f __attribute__((ext_vector_type(2)));
typedef float v8f __attribute__((ext_vector_type(8)));

struct C2 { float re, im; };
__device__ __forceinline__ C2 cmul(C2 a, C2 b) {
    return { a.re * b.re - a.im * b.im, a.re * b.im + a.im * b.re };
}

// ---------------------------------------------------------------------------
// Kernel 1: per-(h,n) setup.  Computes
//   dt   = exp(log_dt[h])
//   dtA  = (-exp(inv_A_real) + i*A_imag) * dt
//   Cm   = 2 * (B*C) * dt / (1 - dtA/2)      (factor 2 from 2*Re(...) folded in)
//   w    = exp(dtA)        (|w| < 1, decaying mode)
//   W    = w^16
// into d_ws as 6 planes of H*N floats.
// ---------------------------------------------------------------------------
__global__ void s4d_setup(const float* __restrict__ log_dt,
                          const float* __restrict__ C_re,
                          const float* __restrict__ C_im,
                          const float* __restrict__ B_re,
                          const float* __restrict__ B_im,
                          const float* __restrict__ inv_A_real,
                          const float* __restrict__ A_imag,
                          float* __restrict__ ws) {
    int idx = blockIdx.x * blockDim.x + threadIdx.x;   // h*N + n
    if (idx >= H_ * N_) return;
    int h = idx / N_;
    int n = idx - h * N_;
    int hs = h % NSSM_;                                 // tile over n_ssm (rep)

    float dt  = __expf(log_dt[h]);
    float ar  = -__expf(inv_A_real[hs * N_ + n]);
    float ai  = A_imag[hs * N_ + n];
    float dre = ar * dt;                                // Re(dtA)  (always < 0)
    float dim = ai * dt;                                // Im(dtA)

    // fac = dt / (1 - dtA/2)
    float denr = 1.0f - 0.5f * dre;
    float deni = -0.5f * dim;
    float s    = dt / (denr * denr + deni * deni);
    C2 fac = { s * denr, -s * deni };

    C2 Bv = { B_re[hs * N_ + n], B_im[hs * N_ + n] };
    C2 Cv = { C_re[h * N_ + n],  C_im[h * N_ + n]  };   // channels == 1
    C2 cm = cmul(cmul(Bv, Cv), fac);
    cm.re *= 2.0f; cm.im *= 2.0f;

    // w = exp(dtA)
    float ew = __expf(dre);
    float sn, cs;
    __sincosf(dim, &sn, &cs);
    C2 w = { ew * cs, ew * sn };
    C2 W = w;                                           // w^16 by 4 squarings
    W = cmul(W, W); W = cmul(W, W); W = cmul(W, W); W = cmul(W, W);

    const int P = H_ * N_;
    ws[0 * P + idx] = cm.re;
    ws[1 * P + idx] = cm.im;
    ws[2 * P + idx] = w.re;
    ws[3 * P + idx] = w.im;
    ws[4 * P + idx] = W.re;
    ws[5 * P + idx] = W.im;
}

// ---------------------------------------------------------------------------
// Kernel 2: Vandermonde GEMM via WMMA.
//   l = 16*lhi + llo ;  K[h,l] = Re( P[lhi,:] . Q[:,llo] )
//   P[lhi,n] = Cm_n * W_n^lhi   (A-matrix, M=lhi rows, K=n)
//   Q[n,llo] = w_n^llo          (B-matrix, K=n rows,  N=llo)
//   D = P_re*Q_re + (-P_im)*Q_im   (imag plane stored negated in LDS,
//                                   since f32 WMMA has no A-negate modifier)
// Block = (h, mhalf): 4 waves, each wave = one 16x16 D tile chain of
// 16 K-steps x 2 wmma_f32_16x16x4_f32 accumulations.
// ---------------------------------------------------------------------------
__global__ void __launch_bounds__(128) s4d_vand(const float* __restrict__ ws,
                                                float* __restrict__ out) {
    __shared__ __align__(16) float Pre[PROWS * PSTRIDE];
    __shared__ __align__(16) float Pim[PROWS * PSTRIDE];   // holds -Im(P)
    __shared__ __align__(16) float Qre[N_ * 16];            // pair-interleaved over k
    __shared__ __align__(16) float Qim[N_ * 16];

    const int h     = blockIdx.x >> 1;
    const int mhalf = blockIdx.x & 1;        // which 64 lhi rows
    const int tid   = threadIdx.x;

    const int P = H_ * N_;
    const int k = tid & 63;                  // mode index
    const int g = tid >> 6;                  // 0..1 : row group of 32

    C2 cm = { ws[0 * P + h * N_ + k], ws[1 * P + h * N_ + k] };
    C2 w  = { ws[2 * P + h * N_ + k], ws[3 * P + h * N_ + k] };
    C2 W  = { ws[4 * P + h * N_ + k], ws[5 * P + h * N_ + k] };

    // ---- Q generation (threads 0..63): Q[k][llo] = w^llo, llo = 0..15,
    // stored pair-interleaved: Q[((k/2)*16 + llo)*2 + (k&1)] so a lane's
    // B fragment {k, k+1} is one aligned b64 LDS read.
    if (tid < 64) {
        C2 p = { 1.0f, 0.0f };
        const int base = ((k >> 1) * 16) * 2 + (k & 1);
        #pragma unroll
        for (int l = 0; l < 16; ++l) {
            Qre[base + 2 * l] = p.re;
            Qim[base + 2 * l] = p.im;
            p = cmul(p, w);
        }
    }

    // ---- P generation: local rows r = 32*g .. 32*g+31 map to
    // global lhi = 64*mhalf + r.   P = Cm * W^lhi  via recurrence.
    {
        C2 E = W;                            // -> W^32 by 5 squarings
        E = cmul(E, E); E = cmul(E, E); E = cmul(E, E);
        E = cmul(E, E); E = cmul(E, E);
        const int sexp = 2 * mhalf + g;      // base = W^(32*sexp)
        C2 base = { 1.0f, 0.0f };
        for (int t = 0; t < sexp; ++t) base = cmul(base, E);
        C2 p = cmul(cm, base);
        const int r0 = 32 * g;
        for (int i = 0; i < 32; ++i) {
            Pre[(r0 + i) * PSTRIDE + k] =  p.re;
            Pim[(r0 + i) * PSTRIDE + k] = -p.im;
            p = cmul(p, W);
        }
    }
    __syncthreads();

    // ---- WMMA phase: wave -> 16-row M tile.
    const int wave = tid >> 5;               // 0..3
    const int lane = tid & 31;
    const int mrow = (wave << 4) + (lane & 15);   // local A row (M = lane&15)
    const int half = lane >> 4;                   // K offset select (+2)
    const int ncol = lane & 15;

    v8f acc = { 0.f, 0.f, 0.f, 0.f, 0.f, 0.f, 0.f, 0.f };

    #pragma unroll
    for (int kk = 0; kk < 16; ++kk) {
        const int kA = 4 * kk + 2 * half;    // this lane's K pair {kA, kA+1}
        v2f a_re = *(const v2f*)&Pre[mrow * PSTRIDE + kA];
        v2f a_im = *(const v2f*)&Pim[mrow * PSTRIDE + kA];
        const int qoff = ((2 * kk + half) * 16 + ncol) * 2;
        v2f b_re = *(const v2f*)&Qre[qoff];
        v2f b_im = *(const v2f*)&Qim[qoff];
        // D += P_re * Q_re ; D += (-P_im) * Q_im
        acc = __builtin_amdgcn_wmma_f32_16x16x4_f32(
                  false, a_re, false, b_re, (short)0, acc, false, false);
        acc = __builtin_amdgcn_wmma_f32_16x16x4_f32(
                  false, a_im, false, b_im, (short)0, acc, false, false);
    }

    // ---- Store: D vgpr j -> M = j + 8*half (within tile), N = ncol.
    // l = lhi*16 + llo : lanes 0..15 of each j write 64B contiguous.
    float* o = out + h * L_;
    const int lhi_base = 64 * mhalf + (wave << 4);
    #pragma unroll
    for (int j = 0; j < 8; ++j) {
        const int lhi = lhi_base + j + 8 * half;
        o[lhi * 16 + ncol] = acc[j];
    }
}

// ---------------------------------------------------------------------------
extern "C" void kernel_launch(void* const* d_in, const int* in_sizes, int n_in,
                              void* d_out, int out_size, void* d_ws, size_t ws_size,
                              hipStream_t stream) {
    // inputs: [0]=L (scalar), [1]=log_dt, [2]=C_re, [3]=C_im,
    //         [4]=B_re, [5]=B_im, [6]=inv_A_real, [7]=A_imag
    const float* log_dt = (const float*)d_in[1];
    const float* C_re   = (const float*)d_in[2];
    const float* C_im   = (const float*)d_in[3];
    const float* B_re   = (const float*)d_in[4];
    const float* B_im   = (const float*)d_in[5];
    const float* inv_A  = (const float*)d_in[6];
    const float* A_im   = (const float*)d_in[7];
    float* ws  = (float*)d_ws;     // needs 6*H*N floats = 1.5 MB
    float* out = (float*)d_out;    // H*L floats

    s4d_setup<<<(H_ * N_ + 255) / 256, 256, 0, stream>>>(
        log_dt, C_re, C_im, B_re, B_im, inv_A, A_im, ws);
    s4d_vand<<<H_ * 2, 128, 0, stream>>>(ws, out);
}